// VectorQuantizer_29033978921144
// MI455X (gfx1250) — compile-verified
//
#include <hip/hip_runtime.h>
#include <hip/hip_bf16.h>

// VQ-VAE vector quantizer for MI455X (gfx1250, wave32, WMMA f32 16x16x4).
// d_in[0] = z (65536x256 f32), d_in[1] = codebook (1024x256 f32)
// d_out   = [z_q_st (65536*256 f32), vq_loss, perplexity]

#define B_ROWS   65536
#define DIM      256
#define K_CODES  1024
#define WAVES    4
#define BLK      (WAVES * 32)
#define ROW_TILE (WAVES * 16)     // 64 z-rows per block, 16 per wave
#define NTILE    32               // codes per LDS tile (2 WMMA N-tiles)
#define CSTRIDE  260              // padded LDS row stride: 260 % 64 = 4 -> conflict-free frags

typedef float v2f __attribute__((ext_vector_type(2)));
typedef float v8f __attribute__((ext_vector_type(8)));

// ---------------- init: zero counts + mse accumulator ----------------
__global__ void vq_init(unsigned* __restrict__ counts, float* __restrict__ mse) {
    int i = blockIdx.x * blockDim.x + threadIdx.x;
    if (i < K_CODES) counts[i] = 0u;
    if (i == 0) *mse = 0.0f;
}

// ---------------- ||e_k||^2 per codebook row (one wave per row) ----------------
__global__ void vq_esq(const float* __restrict__ cb, float* __restrict__ esq) {
    int wave = threadIdx.x >> 5;
    int lane = threadIdx.x & 31;
    int row  = blockIdx.x * (blockDim.x >> 5) + wave;
    if (row >= K_CODES) return;
    const float* r = cb + (size_t)row * DIM;
    float s = 0.0f;
    for (int c = lane; c < DIM; c += 32) { float v = r[c]; s += v * v; }
    #pragma unroll
    for (int off = 16; off; off >>= 1) s += __shfl_xor(s, off, 32);
    if (lane == 0) esq[row] = s;
}

// ---------------- fused GEMM(WMMA f32) + argmin + gather + losses ----------------
__global__ __launch_bounds__(BLK) void vq_main(
    const float* __restrict__ z, const float* __restrict__ cbk,
    const float* __restrict__ esq, unsigned* __restrict__ counts,
    float* __restrict__ msesum, float* __restrict__ out)
{
    __shared__ float cbt[NTILE * CSTRIDE];     // 32-code codebook tile (~33 KB)
    __shared__ int   idxb[WAVES][16];          // per-row argmin indices

    const int tid     = threadIdx.x;
    const int lane    = tid & 31;
    const int wave    = tid >> 5;
    const int rowBase = blockIdx.x * ROW_TILE;

    // A fragment mapping (16x4 f32, ISA 7.12.2): lane&15 -> M row,
    // lane>>4 selects K offset {0,1} vs {2,3} within each 4-wide K step.
    const int mrow  = 16 * wave + (lane & 15);
    const int khalf = (lane >> 4) * 2;
    const int ncol  = lane & 15;

    // Preload this wave's 16x256 A block into registers: 64 k-step fragments,
    // 2 VGPRs each = 128 VGPRs. Loaded once, reused for all 1024 codes.
    v2f afrag[DIM / 4];
    {
        const float* zr = z + (size_t)(rowBase + mrow) * DIM + khalf;
        #pragma unroll
        for (int s = 0; s < DIM / 4; ++s)
            afrag[s] = *(const v2f*)(zr + 4 * s);
    }

    // Running per-lane argmin. C layout: VGPR r, lanes 0-15 = (M=r, N=lane),
    // lanes 16-31 = (M=r+8, N=lane-16).
    float best[8]; int bidx[8];
    #pragma unroll
    for (int r = 0; r < 8; ++r) { best[r] = 3.4e38f; bidx[r] = 0; }

    for (int nt = 0; nt < K_CODES / NTILE; ++nt) {
        __syncthreads();  // previous tile's reads done
        for (int i = tid; i < NTILE * (DIM / 4); i += BLK) {
            int r  = i / (DIM / 4);
            int c4 = i % (DIM / 4);
            float4 v = ((const float4*)(cbk + (size_t)(nt * NTILE + r) * DIM))[c4];
            *(float4*)&cbt[r * CSTRIDE + c4 * 4] = v;
        }
        __syncthreads();

        // Prefetch next tile into cache while we compute.
        if (nt + 1 < K_CODES / NTILE) {
            const char* nx = (const char*)(cbk + (size_t)(nt + 1) * NTILE * DIM);
            __builtin_prefetch(nx + tid * 256, 0, 1);
        }

        // Two independent accumulator chains over the 32-code tile.
        v8f acc0 = {0.f, 0.f, 0.f, 0.f, 0.f, 0.f, 0.f, 0.f};
        v8f acc1 = {0.f, 0.f, 0.f, 0.f, 0.f, 0.f, 0.f, 0.f};
        #pragma unroll
        for (int s = 0; s < DIM / 4; ++s) {
            // B[k][n] = cbt[n][k] (codebook^T); same half-split as A.
            v2f b0 = *(const v2f*)&cbt[ncol * CSTRIDE        + 4 * s + khalf];
            v2f b1 = *(const v2f*)&cbt[(ncol + 16) * CSTRIDE + 4 * s + khalf];
            acc0 = __builtin_amdgcn_wmma_f32_16x16x4_f32(
                false, afrag[s], false, b0, (short)0, acc0, false, false);
            acc1 = __builtin_amdgcn_wmma_f32_16x16x4_f32(
                false, afrag[s], false, b1, (short)0, acc1, false, false);
        }

        const int   ng0 = nt * NTILE + ncol;
        const int   ng1 = ng0 + 16;
        const float e0  = esq[ng0];
        const float e1  = esq[ng1];
        #pragma unroll
        for (int r = 0; r < 8; ++r) {
            float d0 = e0 - 2.0f * acc0[r];   // ||z||^2 constant per row: dropped
            float d1 = e1 - 2.0f * acc1[r];
            if (d0 < best[r] || (d0 == best[r] && ng0 < bidx[r])) { best[r] = d0; bidx[r] = ng0; }
            if (d1 < best[r] || (d1 == best[r] && ng1 < bidx[r])) { best[r] = d1; bidx[r] = ng1; }
        }
    }

    // Cross-lane argmin within each 16-lane half (xor masks < 16 keep halves apart).
    #pragma unroll
    for (int r = 0; r < 8; ++r) {
        float v = best[r]; int ix = bidx[r];
        #pragma unroll
        for (int off = 1; off < 16; off <<= 1) {
            float ov = __shfl_xor(v, off, 32);
            int   oi = __shfl_xor(ix, off, 32);
            if (ov < v || (ov == v && oi < ix)) { v = ov; ix = oi; }
        }
        if (lane == 0)  { idxb[wave][r]     = ix; atomicAdd(&counts[ix], 1u); }
        if (lane == 16) { idxb[wave][r + 8] = ix; atomicAdd(&counts[ix], 1u); }
    }
    __syncthreads();

    // Epilogue: gather z_q, write z + (z_q - z) (reference fp op order), MSE.
    float ms = 0.0f;
    for (int i = lane; i < 16 * DIM; i += 32) {
        int r = i >> 8;            // DIM == 256
        int c = i & 255;
        int code = idxb[wave][r];
        size_t grow = (size_t)(rowBase + 16 * wave + r) * DIM + c;
        float zq   = cbk[(size_t)code * DIM + c];
        float zv   = z[grow];
        float diff = zq - zv;
        out[grow] = zv + diff;
        ms += diff * diff;
    }
    #pragma unroll
    for (int off = 16; off; off >>= 1) ms += __shfl_xor(ms, off, 32);
    if (lane == 0) atomicAdd(msesum, ms);
}

// ---------------- scalars: perplexity + vq_loss ----------------
__global__ void vq_final(const unsigned* __restrict__ counts,
                         const float* __restrict__ msesum,
                         float* __restrict__ out)
{
    __shared__ float red[256];
    int t = threadIdx.x;
    float s = 0.0f;
    for (int k = t; k < K_CODES; k += 256) {
        float p = (float)counts[k] * (1.0f / (float)B_ROWS);
        s += p * logf(p + 1e-10f);
    }
    red[t] = s; __syncthreads();
    for (int off = 128; off; off >>= 1) {
        if (t < off) red[t] += red[t + off];
        __syncthreads();
    }
    if (t == 0) {
        float perp  = expf(-red[0]);
        float mse   = *msesum * (1.0f / ((float)B_ROWS * (float)DIM));
        float ratio = ((float)K_CODES * 0.1f) / (perp + 1e-10f);
        ratio = fminf(fmaxf(ratio, 0.0f), 10.0f);
        float vq = mse + 0.25f * mse + 0.1f * ratio;
        out[(size_t)B_ROWS * DIM]     = vq;
        out[(size_t)B_ROWS * DIM + 1] = perp;
    }
}

extern "C" void kernel_launch(void* const* d_in, const int* in_sizes, int n_in,
                              void* d_out, int out_size, void* d_ws, size_t ws_size,
                              hipStream_t stream) {
    const float* z   = (const float*)d_in[0];
    const float* cbk = (const float*)d_in[1];
    float* out = (float*)d_out;

    // workspace layout: esq[1024] f32 | counts[1024] u32 | msesum[1] f32
    float*    esq    = (float*)d_ws;
    unsigned* counts = (unsigned*)((char*)d_ws + 4096);
    float*    mse    = (float*)((char*)d_ws + 8192);

    vq_init <<<dim3(5),                 dim3(256), 0, stream>>>(counts, mse);
    vq_esq  <<<dim3(K_CODES / 8),       dim3(256), 0, stream>>>(cbk, esq);
    vq_main <<<dim3(B_ROWS / ROW_TILE), dim3(BLK), 0, stream>>>(z, cbk, esq, counts, mse, out);
    vq_final<<<dim3(1),                 dim3(256), 0, stream>>>(counts, mse, out);
}